// BallQuery_9698036154798
// MI455X (gfx1250) — compile-verified
//
#include <hip/hip_runtime.h>

// Ball query (PointNet++) for MI455X / gfx1250, wave32.
// Distance core via V_WMMA_F32_16X16X4_F32 with the full squared distance
// produced by the matrix unit:
//   A (16x4, points)  row m = (px, py, pz, |p|^2)
//   B (4x16, queries) col n = (-2qx, -2qy, -2qz, 1)
//   C[m,n] = |q_n|^2   (lane-splat accumulator input)
//   D[m,n] = |q|^2 - 2 q.p + |p|^2   ;  sq = max(D, 0)
// Queries ride the N axis: lane n (and n+16) own query n; each lane keeps a
// private top-32 (branchless replace-argmax, register-resident), guarded by a
// min3-tree chunk reject; lane pairs merge via LDS at the end.

typedef __attribute__((ext_vector_type(2))) float v2f;
typedef __attribute__((ext_vector_type(8))) float v8f;

#define NPTS   32768
#define NQ     8192
#define NS     32
#define TILE   2048          // points staged per LDS tile (32 KB as float4)
#define WAVES  4             // 128-thread block = 4 wave32
#define QPW    16            // queries per wave
#define SQ_RAD 0.0625f       // reference masks sqrt-dist < RADIUS^2=0.25 -> sq < 0.0625
#define SENT   3.0e38f

__global__ __launch_bounds__(WAVES * 32)
void ballquery_wmma_kernel(const float* __restrict__ xyz,
                           const float* __restrict__ new_xyz,
                           int* __restrict__ out)
{
    // 32 KB: tile of (x,y,z,|p|^2), reused afterwards as merge scratch
    __shared__ __align__(16) float smem[TILE * 4];

    const int tid  = threadIdx.x;
    const int wave = tid >> 5;
    const int lane = tid & 31;
    const int qi   = lane & 15;      // query slot within wave
    const int half = lane >> 4;      // 0: rows 0-7, 1: rows 8-15 of each chunk

    const int b      = blockIdx.x >> 7;                 // 128 blocks per batch
    const int qBlock = (blockIdx.x & 127) * (WAVES * QPW);
    const int q      = qBlock + wave * QPW + qi;

    // ---- this lane's query -> B operand (4x16 f32 B-matrix layout) ----
    const float* qp = new_xyz + ((size_t)b * NQ + q) * 3;
    const float qx = qp[0], qy = qp[1], qz = qp[2];
    const float q2 = qx * qx + qy * qy + qz * qz;
    v2f bmat;
    bmat.x = half ? (-2.0f * qy) : (-2.0f * qx);   // K=1 : K=0
    bmat.y = half ? 1.0f         : (-2.0f * qz);   // K=3 : K=2
    const v8f cacc = {q2, q2, q2, q2, q2, q2, q2, q2};  // accumulator input

    // ---- per-lane top-32 (covers its half of every 16-pt chunk: 16384 cands)
    float hv[NS]; int hi[NS];
    #pragma unroll
    for (int i = 0; i < NS; ++i) { hv[i] = SENT; hi[i] = 0x7fffffff; }
    float curMax = SENT;             // value of current worst kept candidate
    int   curArg = 0;                // its slot (tracked so insert is branchless)

    const float* xb = xyz + (size_t)b * NPTS * 3;

    for (int tile = 0; tile < NPTS; tile += TILE) {
        __syncthreads();                       // WAR vs previous tile consumers
        for (int k = tid; k < TILE; k += blockDim.x) {
            const float* p = xb + (size_t)(tile + k) * 3;
            const float x = p[0], y = p[1], z = p[2];
            ((float4*)smem)[k] = make_float4(x, y, z, x * x + y * y + z * z);
        }
        __syncthreads();

        // prefetch next tile into L2 while we crunch this one
        if (tile + TILE < NPTS)
            __builtin_prefetch(xb + (size_t)(tile + TILE) * 3 + tid * 24, 0, 1);

        #pragma unroll 2
        for (int tb = 0; tb < TILE; tb += 16) {
            // A operand: lanes 0-15 -> (x,y) of point tb+qi; lanes 16-31 -> (z,|p|^2)
            const v2f av = *(const v2f*)(smem + (tb + qi) * 4 + half * 2);
            v8f c = __builtin_amdgcn_wmma_f32_16x16x4_f32(
                    /*neg_a=*/false, av, /*neg_b=*/false, bmat,
                    /*c_mod=*/(short)0, cacc, /*reuse_a=*/false, /*reuse_b=*/false);

            // cheap reject: min(D) >= curMax  ==>  no clamped sq can pass
            const float m0 = fminf(fminf(c[0], c[1]), fminf(c[2], c[3]));
            const float m1 = fminf(fminf(c[4], c[5]), fminf(c[6], c[7]));
            if (fminf(m0, m1) < curMax) {
                const int pbase = tile + tb + half * 8;
                #pragma unroll
                for (int v = 0; v < 8; ++v) {
                    const float sq = fmaxf(c[v], 0.0f);
                    if (sq < curMax) {         // replace worst, branchlessly
                        const int pidx = pbase + v;
                        #pragma unroll
                        for (int i = 0; i < NS; ++i) {
                            const bool sel = (i == curArg);
                            hv[i] = sel ? sq   : hv[i];
                            hi[i] = sel ? pidx : hi[i];
                        }
                        curMax = hv[0]; curArg = 0;
                        #pragma unroll
                        for (int i = 1; i < NS; ++i) {
                            const bool g = hv[i] > curMax;
                            curMax = g ? hv[i] : curMax;
                            curArg = g ? i     : curArg;
                        }
                    }
                }
            }
        }
    }

    // ---- merge the lane-pair lists per query; emit sorted top-32 ----
    __syncthreads();                           // done with point tiles
    float* vbase = smem + (wave * QPW + qi) * 128;   // 64 vals + 64 idx
    int*   ibase = (int*)(vbase + 64);
    #pragma unroll
    for (int i = 0; i < NS; ++i) {
        vbase[half * NS + i] = hv[i];
        ibase[half * NS + i] = hi[i];
    }
    __syncthreads();

    if (half == 0) {
        const size_t outBase = ((size_t)b * NQ + q) * NS;
        int firstSel = 0;
        for (int k = 0; k < NS; ++k) {         // selection over 64 candidates
            float bm = SENT; int bi = 0x7fffffff; int bj = 0;
            for (int j = 0; j < 2 * NS; ++j) {
                const float v = vbase[j];
                const int  id = ibase[j];
                if (v < bm || (v == bm && id < bi)) { bm = v; bi = id; bj = j; }
            }
            vbase[bj] = SENT;                  // remove selected
            if (k == 0) firstSel = bi;
            out[outBase + k] = (bm < SQ_RAD) ? bi : firstSel;
        }
    }
}

extern "C" void kernel_launch(void* const* d_in, const int* in_sizes, int n_in,
                              void* d_out, int out_size, void* d_ws, size_t ws_size,
                              hipStream_t stream) {
    (void)in_sizes; (void)n_in; (void)out_size; (void)d_ws; (void)ws_size;
    const float* xyz     = (const float*)d_in[0];   // (2, 32768, 3) f32
    const float* new_xyz = (const float*)d_in[1];   // (2, 8192, 3) f32
    int* out = (int*)d_out;                         // (2, 8192, 32) i32

    const int blocksPerBatch = NQ / (WAVES * QPW);  // 128
    dim3 grid(2 * blocksPerBatch);                  // 256
    dim3 block(WAVES * 32);                         // 128 threads = 4 waves
    ballquery_wmma_kernel<<<grid, block, 0, stream>>>(xyz, new_xyz, out);
}